// SquaredEuclideanDistance_24704651886789
// MI455X (gfx1250) — compile-verified
//
#include <hip/hip_runtime.h>

typedef __attribute__((ext_vector_type(2))) float v2f;
typedef __attribute__((ext_vector_type(8))) float v8f;
typedef __attribute__((ext_vector_type(4))) unsigned int u32x4;
typedef __attribute__((ext_vector_type(4))) int i32x4;
typedef __attribute__((ext_vector_type(8))) int i32x8;

#define D_DIM 128
// LDS row stride in dwords: 128 data + 4 pad -> bank = (132*r + c) % 64 = (4r + c) % 64,
// so a ds_load_b64 A/B fragment fetch (lanes: row 0..15, col k0+2*half) touches all
// 64 banks exactly once. Pad is produced by the TDM descriptor itself.
#define LDS_STRIDE 132
#define PANEL_BYTES (128 * LDS_STRIDE * 4)

// ---------------------------------------------------------------------------
// Kernel 1: squared row norms. One wave per row of 128 floats, coalesced
// float4 loads + shuffle reduce.
// ---------------------------------------------------------------------------
__global__ __launch_bounds__(256)
void sed_row_norms(const float* __restrict__ x, float* __restrict__ sq, int rows) {
    const int wave = (blockIdx.x * blockDim.x + threadIdx.x) >> 5;
    const int lane = threadIdx.x & 31;
    if (wave >= rows) return;
    const float4 v = ((const float4*)(x + (size_t)wave * D_DIM))[lane];
    float s = v.x * v.x + v.y * v.y + v.z * v.z + v.w * v.w;
    #pragma unroll
    for (int off = 16; off > 0; off >>= 1)
        s += __shfl_xor(s, off, 32);
    if (lane == 0) sq[wave] = s;
}

// ---------------------------------------------------------------------------
// TDM descriptor build + issue: one tensor_load_to_lds moves a 128x128 f32
// panel (row-major, row stride D_DIM) into LDS with 4-dword row padding.
// D# layout per CDNA5 ISA ch.8 (group0 128b, group1 256b; groups 2/3 unused
// for 2D tiles -> zero-filled).
// ---------------------------------------------------------------------------
__device__ __forceinline__ void tdm_load_panel(const float* gptr, unsigned int lds_off) {
    const unsigned long long ga = (unsigned long long)(uintptr_t)gptr;

    u32x4 g0;
    g0.x = 1u;                                        // count=1, user mode, no gather
    g0.y = lds_off;                                   // lds_addr [63:32]
    g0.z = (unsigned int)(ga & 0xFFFFFFFFu);          // global_addr [95:64]
    g0.w = (unsigned int)((ga >> 32) & 0x1FFFFFFu)    // global_addr [120:96]
         | (2u << 30);                                // type=2 ("image") [127:126]

    i32x8 g1;
    g1[0] = (int)((2u << 16)      // data_size = 4 bytes
                | (1u << 20)      // pad_enable
                | (6u << 22)      // pad_interval: every 128 dwords
                | (3u << 25));    // pad_amount: 4 dwords
    g1[1] = (int)(128u << 16);    // tensor_dim0 = 128 (bits 79:48)
    g1[2] = (int)(128u << 16);    // tensor_dim0_hi=0, tensor_dim1 = 128 (bits 111:80)
    g1[3] = (int)(128u << 16);    // tensor_dim1_hi=0, tile_dim0 = 128 (bits 127:112)
    g1[4] = 128;                  // tile_dim1 = 128, tile_dim2 = 0
    g1[5] = D_DIM;                // tensor_dim0_stride = 128 elements
    g1[6] = 0;                    // stride0_hi = 0, tensor_dim1_stride lo = 0
    g1[7] = 0;

    const i32x4 z4 = {0, 0, 0, 0};
#if defined(__clang_major__) && __clang_major__ >= 23
    const i32x8 z8 = {0, 0, 0, 0, 0, 0, 0, 0};
    __builtin_amdgcn_tensor_load_to_lds(g0, g1, z4, z4, z8, 0);
#else
    __builtin_amdgcn_tensor_load_to_lds(g0, g1, z4, z4, 0);
#endif
}

// ---------------------------------------------------------------------------
// Kernel 2: 128x128 output tile per 256-thread block (8 waves).
// TDM stages both input panels into padded LDS; each wave computes a
// 32(N) x 64(M) region = 2x4 WMMA f32 16x16x4 tiles, K fully unrolled.
// ---------------------------------------------------------------------------
__global__ __launch_bounds__(256)
void sed_wmma_kernel(const float* __restrict__ lhs, const float* __restrict__ rhs,
                     const float* __restrict__ lsq, const float* __restrict__ rsq,
                     float* __restrict__ out, int M) {
    __shared__ float smem[2 * 128 * LDS_STRIDE];
    float* lhsS = smem;
    float* rhsS = smem + 128 * LDS_STRIDE;

    const int lane = threadIdx.x & 31;
    const int wave = threadIdx.x >> 5;
    const int wN = wave >> 1;   // 0..3
    const int wM = wave & 1;    // 0..1
    const int lr = lane & 15;   // fragment row (A) / col (B)
    const int lh = lane >> 4;   // K-half selector

    const int n_blk = blockIdx.y * 128;
    const int m_blk = blockIdx.x * 128;

    // --- Stage both panels via the Tensor Data Mover (wave 0 only) ---
    if (wave == 0) {
        const unsigned int lds_base = (unsigned int)(uintptr_t)smem; // addr[31:0] == LDS offset
        tdm_load_panel(lhs + (size_t)n_blk * D_DIM, lds_base);
        tdm_load_panel(rhs + (size_t)m_blk * D_DIM, lds_base + PANEL_BYTES);
        __builtin_amdgcn_s_wait_tensorcnt(0);
    }
    __syncthreads();

    // --- Fragment base pointers into padded LDS ---
    const float* aptr[2];
    const float* bptr[4];
    #pragma unroll
    for (int t = 0; t < 2; ++t)
        aptr[t] = lhsS + (wN * 32 + t * 16 + lr) * LDS_STRIDE + lh * 2;
    #pragma unroll
    for (int t = 0; t < 4; ++t)
        bptr[t] = rhsS + (wM * 64 + t * 16 + lr) * LDS_STRIDE + lh * 2;

    v8f acc[2][4];
    #pragma unroll
    for (int i = 0; i < 2; ++i)
        #pragma unroll
        for (int j = 0; j < 4; ++j)
            acc[i][j] = 0.0f;

    // K loop: 32 steps of K=4 (256 WMMAs, 192 conflict-free ds_load_b64).
    #pragma unroll
    for (int k0 = 0; k0 < D_DIM; k0 += 4) {
        v2f a[2], b[4];
        #pragma unroll
        for (int t = 0; t < 2; ++t)
            a[t] = *(const v2f*)(aptr[t] + k0);
        #pragma unroll
        for (int t = 0; t < 4; ++t)
            b[t] = *(const v2f*)(bptr[t] + k0);

        #pragma unroll
        for (int i = 0; i < 2; ++i)
            #pragma unroll
            for (int j = 0; j < 4; ++j)
                acc[i][j] = __builtin_amdgcn_wmma_f32_16x16x4_f32(
                    false, a[i], false, b[j], (short)0, acc[i][j], false, false);
    }

    // --- Epilogue: out = |lhs_row|^2 + |rhs_col|^2 - 2*dot, streamed NT ---
    const int n_base = n_blk + wN * 32;
    const int m_base = m_blk + wM * 64;

    float rv[4];
    #pragma unroll
    for (int j = 0; j < 4; ++j)
        rv[j] = rsq[m_base + j * 16 + lr];

    #pragma unroll
    for (int i = 0; i < 2; ++i) {
        #pragma unroll
        for (int r = 0; r < 8; ++r) {
            const int row = n_base + i * 16 + lh * 8 + r;
            const float lv = lsq[row];
            #pragma unroll
            for (int j = 0; j < 4; ++j) {
                const int col = m_base + j * 16 + lr;
                const float v = lv + rv[j] - 2.0f * acc[i][j][r];
                __builtin_nontemporal_store(v, out + (size_t)row * M + col);
            }
        }
    }
}

extern "C" void kernel_launch(void* const* d_in, const int* in_sizes, int n_in,
                              void* d_out, int out_size, void* d_ws, size_t ws_size,
                              hipStream_t stream) {
    const float* lhs = (const float*)d_in[0];
    const float* rhs = (const float*)d_in[1];
    float* out = (float*)d_out;

    const int N = in_sizes[0] / D_DIM;   // 16384
    const int M = in_sizes[1] / D_DIM;   // 8192

    float* lsq = (float*)d_ws;           // N floats
    float* rsq = lsq + N;                // M floats

    sed_row_norms<<<dim3((N * 32 + 255) / 256), 256, 0, stream>>>(lhs, lsq, N);
    sed_row_norms<<<dim3((M * 32 + 255) / 256), 256, 0, stream>>>(rhs, rsq, M);

    dim3 grid(M / 128, N / 128);         // 64 x 128 blocks
    sed_wmma_kernel<<<grid, 256, 0, stream>>>(lhs, rhs, lsq, rsq, out, M);
}